// PCEN_90787018703374
// MI455X (gfx1250) — compile-verified
//
#include <hip/hip_runtime.h>
#include <cmath>
#include <stdint.h>

// ---------------- problem constants (from reference: B=64, C=80, T=4000) ----
#define EPS 1e-6f
constexpr int T_LEN    = 4000;           // time steps per row
constexpr int LANES    = 32;             // wave32
constexpr int CHUNK    = T_LEN / LANES;  // 125 elements per lane
constexpr int VEC4     = T_LEN / 4;      // 1000 float4 transfers per row
constexpr int WPB      = 4;              // waves (== rows) per block; LDS = 4*16000B
constexpr int ROW_B    = T_LEN * 4;      // 16000 bytes per row
constexpr int FULL_IT  = VEC4 / LANES;   // 31 full DMA rounds (992 xfers)
constexpr int TAIL_L   = VEC4 - FULL_IT * LANES; // 8 tail lanes

// --- fully-unrolled async DMA bursts: INST_OFFSET advances BOTH global & LDS addr
template <int K, int N>
struct AsyncLoadBurst {
    static __device__ __forceinline__ void run(unsigned laddr, unsigned goff, const float* p) {
        asm volatile("global_load_async_to_lds_b128 %0, %1, %2 offset:%c3"
                     :: "v"(laddr), "v"(goff), "s"(p), "i"(K * LANES * 16) : "memory");
        AsyncLoadBurst<K + 1, N>::run(laddr, goff, p);
    }
};
template <int N>
struct AsyncLoadBurst<N, N> {
    static __device__ __forceinline__ void run(unsigned, unsigned, const float*) {}
};

template <int K, int N>
struct AsyncStoreBurst {
    static __device__ __forceinline__ void run(unsigned laddr, unsigned goff, float* p) {
        asm volatile("global_store_async_from_lds_b128 %0, %1, %2 offset:%c3"
                     :: "v"(goff), "v"(laddr), "s"(p), "i"(K * LANES * 16) : "memory");
        AsyncStoreBurst<K + 1, N>::run(laddr, goff, p);
    }
};
template <int N>
struct AsyncStoreBurst<N, N> {
    static __device__ __forceinline__ void run(unsigned, unsigned, float*) {}
};

// PCEN: M_t = (1-s) M_{t-1} + s E_t ;  out = (E/(eps+M)^alpha + delta)^r - delta^r
extern "C" __global__ __launch_bounds__(LANES * WPB)
void pcen_kernel(const float* __restrict__ E,
                 const float* __restrict__ log_alpha,
                 const float* __restrict__ log_delta,
                 const float* __restrict__ log_r,
                 const float* __restrict__ log_s,
                 float* __restrict__ out,
                 int C, int rows)
{
    __shared__ __align__(16) float sh[WPB][T_LEN];

    const int lane = threadIdx.x & (LANES - 1);
    const int wv   = threadIdx.x >> 5;                // wave id in block
    const int row  = blockIdx.x * WPB + wv;           // row = b*C + c
    if (row >= rows) return;                          // whole wave exits (no barriers used)
    const int c = row % C;

    float* const my = sh[wv];

    // 32-bit byte offsets (max ~82MB < 2^31): base pointers stay uniform in SGPRs
    const unsigned goff  = (unsigned)row * (unsigned)ROW_B + (unsigned)(lane * 16);
    const unsigned laddr = (unsigned)(uintptr_t)(&my[lane * 4]);

    // ---- Stage 1: async DMA row -> LDS: 31 full bursts + 8-lane tail --------
    AsyncLoadBurst<0, FULL_IT>::run(laddr, goff, E);
    if (lane < TAIL_L) {
        asm volatile("global_load_async_to_lds_b128 %0, %1, %2 offset:%c3"
                     :: "v"(laddr), "v"(goff), "s"(E), "i"(FULL_IT * LANES * 16) : "memory");
    }
    asm volatile("s_wait_asynccnt 0x0" ::: "memory");   // row resident in LDS

    // ---- Per-channel parameters (computed redundantly per lane, cheap) ------
    const float alpha = 1.0f / (1.0f + __expf(-log_alpha[c]));
    const float xd    = log_delta[c];
    const float delta = (xd > 20.0f) ? xd : log1pf(__expf(xd));    // softplus
    const float r     = 1.0f / (1.0f + __expf(-log_r[c]));
    const float s     = 1.0f / (1.0f + __expf(-log_s[c]));
    const float a     = 1.0f - s;
    const float dr    = exp2f(r * log2f(delta));                   // delta^r

    const int t0 = lane * CHUNK;   // lane's chunk [t0, t0+CHUNK)
    // LDS banks: lane stride = 125 floats -> bank step 61 (coprime with 64): conflict-free

    // ---- Pass 1: local EMA with zero initial state -> chunk-end value -------
    float m = 0.0f;
    #pragma unroll 5
    for (int i = 0; i < CHUNK; ++i)
        m = fmaf(a, m, s * my[t0 + i]);

    // ---- Cross-lane linear-recurrence scan of chunk carries -----------------
    // c_k = a^CHUNK * c_{k-1} + end_k ; Hillis-Steele in 5 shuffle steps (wave32)
    const float qL = exp2f((float)CHUNK * log2f(a));  // a^125
    float v = m, qd = qL;
    #pragma unroll
    for (int d = 1; d < LANES; d <<= 1) {
        float up = __shfl_up(v, (unsigned)d, LANES);
        if (lane >= d) v = fmaf(qd, up, v);
        qd *= qd;                                      // q^d -> q^(2d)
    }
    float carry = __shfl_up(v, 1u, LANES);             // exclusive: state before my chunk
    if (lane == 0) carry = 0.0f;

    // ---- Pass 2: exact rescan + AGC/root compression, result back to LDS ----
    m = carry;
    #pragma unroll 5
    for (int i = 0; i < CHUNK; ++i) {
        float e   = my[t0 + i];
        m         = fmaf(a, m, s * e);
        float den = exp2f(alpha * log2f(EPS + m));     // (eps+M)^alpha, arg > 0
        float x   = e / den + delta;                   // > 0 (delta ~ 2)
        my[t0 + i] = exp2f(r * log2f(x)) - dr;
    }

    // ---- Stage 3: async DMA results LDS -> global, coalesced ----------------
    asm volatile("s_wait_dscnt 0x0" ::: "memory");     // LDS writes landed before DMA reads
    AsyncStoreBurst<0, FULL_IT>::run(laddr, goff, out);
    if (lane < TAIL_L) {
        asm volatile("global_store_async_from_lds_b128 %0, %1, %2 offset:%c3"
                     :: "v"(goff), "v"(laddr), "s"(out), "i"(FULL_IT * LANES * 16) : "memory");
    }
    asm volatile("s_wait_asynccnt 0x0" ::: "memory");  // (S_ENDPGM also waits-idle)
}

extern "C" void kernel_launch(void* const* d_in, const int* in_sizes, int n_in,
                              void* d_out, int out_size, void* d_ws, size_t ws_size,
                              hipStream_t stream) {
    const float* E  = (const float*)d_in[0];
    const float* la = (const float*)d_in[1];
    const float* ld = (const float*)d_in[2];
    const float* lr = (const float*)d_in[3];
    const float* ls = (const float*)d_in[4];
    float* out = (float*)d_out;

    const int C    = in_sizes[1];            // 80
    const int rows = in_sizes[0] / T_LEN;    // B*C = 5120

    dim3 grid((rows + WPB - 1) / WPB);       // 1280 blocks
    dim3 block(LANES * WPB);                 // 128 threads = 4 waves
    pcen_kernel<<<grid, block, 0, stream>>>(E, la, ld, lr, ls, out, C, rows);
}